// SoftDTW_54640573939739
// MI455X (gfx1250) — compile-verified
//
#include <hip/hip_runtime.h>

#define T_LEN 4096
#define DF 256
#define BIGF 1000000000.0f
#define GAMMA_F 0.5f
#define CHUNK 1024

typedef float v2f __attribute__((ext_vector_type(2)));
typedef float v8f __attribute__((ext_vector_type(8)));

// ---------------------------------------------------------------------------
// K0: row squared-norms for x and y. One wave32 per row (256 floats / row,
// 8 floats per lane via two float4 loads, butterfly reduction).
// ---------------------------------------------------------------------------
__global__ void __launch_bounds__(256) sdtw_norms_kernel(
    const float* __restrict__ x, const float* __restrict__ y,
    float* __restrict__ xn, float* __restrict__ yn) {
  int wid  = (blockIdx.x * blockDim.x + threadIdx.x) >> 5;
  int lane = threadIdx.x & 31;
  if (wid >= 2 * T_LEN) return;
  const float* src = (wid < T_LEN) ? x : y;
  int row = (wid < T_LEN) ? wid : wid - T_LEN;
  const float4* p = (const float4*)(src + (size_t)row * DF + lane * 8);
  float4 a = p[0], b = p[1];
  float s = a.x * a.x + a.y * a.y + a.z * a.z + a.w * a.w +
            b.x * b.x + b.y * b.y + b.z * b.z + b.w * b.w;
#pragma unroll
  for (int off = 16; off > 0; off >>= 1) s += __shfl_xor(s, off, 32);
  if (lane == 0) {
    if (wid < T_LEN) xn[row] = s; else yn[row] = s;
  }
}

// ---------------------------------------------------------------------------
// K1: banded pairwise distances via V_WMMA_F32_16X16X4_F32 (full f32).
// One wave32 per 16x16 Gram tile. Tile 2*rb+0 -> columns rb*16-8,
// tile 2*rb+1 -> columns rb*16+8; together they cover all |j-i|<=4 pairs of
// the 16-row block exactly once.
//
// A fragment (16x4 f32, 2 VGPRs/lane): lane L holds A[L&15][2*(L>>4)+v].
// B fragment (4x16 f32) is the mirror: lane L holds B[2*(L>>4)+v][L&15],
// and since B = y^T, that is y[cb + (L&15)][2*(L>>4)+v]  -> same float2
// load pattern as A.
// C/D (16x16 f32, 8 VGPRs): lane L, vgpr v -> element (m = v + 8*(L>>4),
// n = L&15).
// ---------------------------------------------------------------------------
__global__ void __launch_bounds__(32) sdtw_band_wmma_kernel(
    const float* __restrict__ x, const float* __restrict__ y,
    const float* __restrict__ xn, const float* __restrict__ yn,
    float* __restrict__ Dband) {
  int tile = blockIdx.x;                 // 0 .. 511
  int rb   = tile >> 1;                  // 16-row block
  int cb   = rb * 16 + ((tile & 1) ? 8 : -8);
  int lane = threadIdx.x;
  int l15  = lane & 15;
  int hi   = lane >> 4;

  int arow = rb * 16 + l15;                          // always in range
  int brow = cb + l15;
  int brc  = min(max(brow, 0), T_LEN - 1);           // clamp OOB loads

  const float* ap = x + (size_t)arow * DF + 2 * hi;
  const float* bp = y + (size_t)brc  * DF + 2 * hi;

  v8f acc = {};
#pragma unroll 4
  for (int k = 0; k < DF; k += 4) {
    v2f a = *(const v2f*)(ap + k);
    v2f b = *(const v2f*)(bp + k);
    acc = __builtin_amdgcn_wmma_f32_16x16x4_f32(
        /*neg_a=*/false, a, /*neg_b=*/false, b,
        /*c_mod=*/(short)0, acc, /*reuse_a=*/false, /*reuse_b=*/false);
  }

  int bcol = cb + l15;                    // column index of this lane's C cells
  if (bcol >= 0 && bcol < T_LEN) {
    float yb = yn[bcol];
#pragma unroll
    for (int v = 0; v < 8; ++v) {
      int m = v + 8 * hi;
      int a = rb * 16 + m;                // D row index
      int d = bcol - a;                   // diagonal offset
      if (d >= -4 && d <= 4) {
        float d2 = xn[a] + yb - 2.0f * acc[v];
        Dband[(size_t)a * 9 + (d + 4)] = sqrtf(fmaxf(d2, 1e-12f));
      }
    }
  }
}

// ---------------------------------------------------------------------------
// K2: serial banded soft-DTW DP. Single workgroup. All 256 threads stage the
// band buffer into LDS in 36KB chunks; wave 0 runs the 4096-step recurrence
// with the band in lanes 0..8 (slot d = lane-4). r0 = prev slot d,
// r2 = prev slot d+1 (via shfl_down; lanes >= 9 carry BIG). r1 = BIG as in
// the reference. Exact max-subtracted logsumexp, mask 1 <= j <= T.
// ---------------------------------------------------------------------------
__global__ void __launch_bounds__(256) sdtw_dp_kernel(
    const float* __restrict__ Dband, float* __restrict__ out) {
  __shared__ float sD[CHUNK * 9];
  int tid  = threadIdx.x;
  int lane = tid & 31;

  // Row 0 (= R0): slot d holds R0[j=d]; only j=0 is 0, rest BIG.
  float prevR = (tid == 4) ? 0.0f : BIGF;
  const float inv_g = 1.0f / GAMMA_F;    // 2.0
  const float s1    = -BIGF * inv_g;     // the disabled r1=BIG term

  for (int c = 0; c < T_LEN / CHUNK; ++c) {
    for (int idx = tid; idx < CHUNK * 9; idx += 256)
      sD[idx] = Dband[(size_t)c * CHUNK * 9 + idx];
    __syncthreads();

    if (tid < 32) {
      for (int r = 0; r < CHUNK; ++r) {
        int i = c * CHUNK + r + 1;                 // DP row (1-based)
        float r0 = prevR;                          // R_prev[j-1] (slot d)
        float r2 = __shfl_down(prevR, 1, 32);      // R_prev[j]   (slot d+1)
        float s0 = -r0 * inv_g;
        float s2 = -r2 * inv_g;
        float mx = fmaxf(fmaxf(s0, s2), s1);
        float sum = __expf(s0 - mx) + __expf(s1 - mx) + __expf(s2 - mx);
        float smin = -GAMMA_F * (mx + __logf(sum));
        float nR = BIGF;
        if (lane < 9) {
          int j = i + lane - 4;
          if (j >= 1 && j <= T_LEN) nR = sD[r * 9 + lane] + smin;
        }
        prevR = nR;
      }
    }
    __syncthreads();
  }
  // Final answer: R[T][T] -> row i=T, d=0 -> lane 4.
  if (tid == 4) out[0] = prevR;
}

// ---------------------------------------------------------------------------
extern "C" void kernel_launch(void* const* d_in, const int* in_sizes, int n_in,
                              void* d_out, int out_size, void* d_ws, size_t ws_size,
                              hipStream_t stream) {
  (void)in_sizes; (void)n_in; (void)out_size; (void)ws_size;
  const float* x = (const float*)d_in[0];
  const float* y = (const float*)d_in[1];

  float* ws    = (float*)d_ws;
  float* Dband = ws;                 // T*9 floats = 147456 B
  float* xn    = ws + T_LEN * 9;     // T floats
  float* yn    = xn + T_LEN;         // T floats

  sdtw_norms_kernel<<<(2 * T_LEN) / 8, 256, 0, stream>>>(x, y, xn, yn);
  sdtw_band_wmma_kernel<<<2 * (T_LEN / 16), 32, 0, stream>>>(x, y, xn, yn, Dband);
  sdtw_dp_kernel<<<1, 256, 0, stream>>>(Dband, (float*)d_out);
}